// VQuantLinear_52596169507334
// MI455X (gfx1250) — compile-verified
//
#include <hip/hip_runtime.h>

typedef __attribute__((ext_vector_type(2))) float v2f;
typedef __attribute__((ext_vector_type(8))) float v8f;

// Problem constants
#define IN_F     4096
#define OUT_F    4096
#define M_TOT    8192          // 2 * 4096 rows of x
#define V_LEN    8
#define OV_LEN   4
#define GS       3968
#define NI       512
#define ONI      1024
#define OUTLIER  128

// GEMM tiling
#define BM  128
#define BN  128
#define BK  32
#define BKP 34                 // +2 pad: keeps ds_load_b64 8B-aligned, kills bank conflicts

// ---------------------------------------------------------------------------
// Kernel 1: dequantize weight into Wt[c][o] = W_final[o][c]  (c = perm[k])
//   main:    raw[o, 128+j] = cb[idx[o/8, j]][o%8] + rcb[ridx[o/8, j]][o%8]
//   outlier: raw[o, k]     = ocb[oidx[o/4, k]][o%4]          (k < 128)
//   Wt[perm[k]][o] = raw[o,k] * scale[k] + bias[k]
// One wave per column k; lanes sweep o. Column writes fully coalesced.
// ---------------------------------------------------------------------------
__global__ __launch_bounds__(256) void vq_dequant(
    const float* __restrict__ centroids,
    const float* __restrict__ res_centroids,
    const float* __restrict__ outlier_centroids,
    const float* __restrict__ wscale,
    const float* __restrict__ wbias,
    const int*   __restrict__ indices,
    const int*   __restrict__ res_indices,
    const int*   __restrict__ outlier_indices,
    const int*   __restrict__ perm,
    float*       __restrict__ Wt)
{
    const int wave = threadIdx.x >> 5;
    const int lane = threadIdx.x & 31;
    const int k = blockIdx.x * 8 + wave;          // quantized column 0..4095
    const float sc = wscale[k];
    const float bi = wbias[k];
    const int   c  = perm[k];                     // final in_feature position
    float* __restrict__ col = Wt + (size_t)c * (size_t)OUT_F;

    if (k < OUTLIER) {
        // outlier columns: o = a*4 + v
        for (int a = lane; a < ONI; a += 32) {
            const int oi = outlier_indices[a * OUTLIER + k];
            const float4 cv = *(const float4*)(outlier_centroids + oi * OV_LEN);
            float4 r;
            r.x = cv.x * sc + bi;
            r.y = cv.y * sc + bi;
            r.z = cv.z * sc + bi;
            r.w = cv.w * sc + bi;
            *(float4*)(col + a * 4) = r;
        }
    } else {
        const int j = k - OUTLIER;                // column within main group
        for (int n = lane; n < NI; n += 32) {
            const int idx  = indices[n * GS + j];
            const int ridx = res_indices[n * GS + j];
            const float4* cv = (const float4*)(centroids     + idx  * V_LEN);
            const float4* rv = (const float4*)(res_centroids + ridx * V_LEN);
            const float4 c0 = cv[0], c1 = cv[1];
            const float4 r0 = rv[0], r1 = rv[1];
            float4 o0, o1;
            o0.x = (c0.x + r0.x) * sc + bi;
            o0.y = (c0.y + r0.y) * sc + bi;
            o0.z = (c0.z + r0.z) * sc + bi;
            o0.w = (c0.w + r0.w) * sc + bi;
            o1.x = (c1.x + r1.x) * sc + bi;
            o1.y = (c1.y + r1.y) * sc + bi;
            o1.z = (c1.z + r1.z) * sc + bi;
            o1.w = (c1.w + r1.w) * sc + bi;
            *(float4*)(col + n * 8)     = o0;
            *(float4*)(col + n * 8 + 4) = o1;
        }
    }
}

// ---------------------------------------------------------------------------
// Kernel 2: C(8192x4096) = A(8192x4096) * B(4096x4096), all row-major f32.
// Block tile 128x128x32, 256 threads = 8 waves (wave32), wave tile 32x64,
// accumulated with v_wmma_f32_16x16x4_f32 (2x4 fragments of 16x16 per wave).
//
// Fragment layouts (gfx1250, wave32):
//   A 16x4:  vgpr0 = A[m][kk + 2*half], vgpr1 = A[m][kk + 2*half + 1], m = lane%16
//   B 4x16:  vgpr0 = B[kk + 2*half][n], vgpr1 = B[kk + 2*half + 1][n], n = lane%16
//   C 16x16: vgpr v -> (M = v + 8*half, N = lane%16)
// LDS tiles stored k-fastest (As[m][kk], Bs[n][kk]) so each fragment half is
// one aligned 8-byte LDS load per matrix tile.
// ---------------------------------------------------------------------------
__global__ __launch_bounds__(256) void vq_gemm(
    const float* __restrict__ A,
    const float* __restrict__ B,
    float*       __restrict__ C)
{
    __shared__ float As[BM][BKP];
    __shared__ float Bs[BN][BKP];

    const int t    = threadIdx.x;
    const int lane = t & 31;
    const int wave = t >> 5;
    const int half = lane >> 4;     // 0: lanes 0-15, 1: lanes 16-31
    const int lm   = lane & 15;
    const int wm   = wave & 3;      // wave M index (4)
    const int wn   = wave >> 2;     // wave N index (2)

    const int mBase = blockIdx.y * BM;
    const int nBase = blockIdx.x * BN;

    // global->LDS load mapping
    const int ar  = t >> 3;          // A row in tile (0..31, step 32)
    const int ac  = (t & 7) << 2;    // A k in tile   (0..28, float4)
    const int bc  = (t & 31) << 2;   // B col in tile (0..124, float4)
    const int bkr = t >> 5;          // B k row in tile (0..7, step 8)

    v8f acc[2][4];
#pragma unroll
    for (int i = 0; i < 2; ++i)
#pragma unroll
        for (int j = 0; j < 4; ++j)
            acc[i][j] = (v8f){0.f, 0.f, 0.f, 0.f, 0.f, 0.f, 0.f, 0.f};

    float4 aReg[4], bReg[4];
#pragma unroll
    for (int i = 0; i < 4; ++i)
        aReg[i] = *(const float4*)(A + (size_t)(mBase + ar + 32 * i) * IN_F + ac);
#pragma unroll
    for (int i = 0; i < 4; ++i)
        bReg[i] = *(const float4*)(B + (size_t)(bkr + 8 * i) * OUT_F + nBase + bc);

    for (int kt = 0; kt < IN_F; kt += BK) {
        __syncthreads();
        // stage tiles into LDS (A: k-fastest direct; B: transposed n-major)
#pragma unroll
        for (int i = 0; i < 4; ++i) {
            const float4 v = aReg[i];
            float* p = &As[ar + 32 * i][ac];
            p[0] = v.x; p[1] = v.y; p[2] = v.z; p[3] = v.w;
        }
#pragma unroll
        for (int i = 0; i < 4; ++i) {
            const float4 v = bReg[i];
            const int kr = bkr + 8 * i;
            Bs[bc + 0][kr] = v.x;
            Bs[bc + 1][kr] = v.y;
            Bs[bc + 2][kr] = v.z;
            Bs[bc + 3][kr] = v.w;
        }
        __syncthreads();

        // issue next tile's global loads before compute (latency overlap)
        const int ktn = kt + BK;
        if (ktn < IN_F) {
#pragma unroll
            for (int i = 0; i < 4; ++i)
                aReg[i] = *(const float4*)(A + (size_t)(mBase + ar + 32 * i) * IN_F + ktn + ac);
#pragma unroll
            for (int i = 0; i < 4; ++i)
                bReg[i] = *(const float4*)(B + (size_t)(ktn + bkr + 8 * i) * OUT_F + nBase + bc);
        }

        // 8 k-steps x 8 WMMA = 64 v_wmma_f32_16x16x4_f32 per tile iteration
#pragma unroll
        for (int kk = 0; kk < BK; kk += 4) {
            const int kA = kk + half * 2;
            const v2f a0 = *(const v2f*)&As[wm * 32 +      lm][kA];
            const v2f a1 = *(const v2f*)&As[wm * 32 + 16 + lm][kA];
            const v2f b0 = *(const v2f*)&Bs[wn * 64 +      lm][kA];
            const v2f b1 = *(const v2f*)&Bs[wn * 64 + 16 + lm][kA];
            const v2f b2 = *(const v2f*)&Bs[wn * 64 + 32 + lm][kA];
            const v2f b3 = *(const v2f*)&Bs[wn * 64 + 48 + lm][kA];
            acc[0][0] = __builtin_amdgcn_wmma_f32_16x16x4_f32(false, a0, false, b0, (short)0, acc[0][0], false, false);
            acc[0][1] = __builtin_amdgcn_wmma_f32_16x16x4_f32(false, a0, false, b1, (short)0, acc[0][1], false, false);
            acc[0][2] = __builtin_amdgcn_wmma_f32_16x16x4_f32(false, a0, false, b2, (short)0, acc[0][2], false, false);
            acc[0][3] = __builtin_amdgcn_wmma_f32_16x16x4_f32(false, a0, false, b3, (short)0, acc[0][3], false, false);
            acc[1][0] = __builtin_amdgcn_wmma_f32_16x16x4_f32(false, a1, false, b0, (short)0, acc[1][0], false, false);
            acc[1][1] = __builtin_amdgcn_wmma_f32_16x16x4_f32(false, a1, false, b1, (short)0, acc[1][1], false, false);
            acc[1][2] = __builtin_amdgcn_wmma_f32_16x16x4_f32(false, a1, false, b2, (short)0, acc[1][2], false, false);
            acc[1][3] = __builtin_amdgcn_wmma_f32_16x16x4_f32(false, a1, false, b3, (short)0, acc[1][3], false, false);
        }
    }

    // epilogue: acc vgpr v -> row (base + v + 8*half), col (base + lane%16)
    const int rBase = mBase + wm * 32;
    const int cBase = nBase + wn * 64;
#pragma unroll
    for (int i = 0; i < 2; ++i) {
        const int row0 = rBase + i * 16 + half * 8;
#pragma unroll
        for (int j = 0; j < 4; ++j) {
            const int colI = cBase + j * 16 + lm;
#pragma unroll
            for (int v = 0; v < 8; ++v)
                C[(size_t)(row0 + v) * OUT_F + colI] = acc[i][j][v];
        }
    }
}

// ---------------------------------------------------------------------------
extern "C" void kernel_launch(void* const* d_in, const int* in_sizes, int n_in,
                              void* d_out, int out_size, void* d_ws, size_t ws_size,
                              hipStream_t stream)
{
    const float* x                 = (const float*)d_in[0];
    const float* centroids         = (const float*)d_in[1];
    const float* res_centroids     = (const float*)d_in[2];
    const float* outlier_centroids = (const float*)d_in[3];
    const float* wscale            = (const float*)d_in[4];
    const float* wbias             = (const float*)d_in[5];
    const int*   indices           = (const int*)d_in[6];
    const int*   res_indices       = (const int*)d_in[7];
    const int*   outlier_indices   = (const int*)d_in[8];
    const int*   perm              = (const int*)d_in[9];
    float*       out               = (float*)d_out;

    float* Wt = (float*)d_ws;   // 4096 x 4096 f32 = 64 MB scratch (fits in L2)

    // Phase 1: dequantize + scale/bias + perm-scatter into transposed weight
    vq_dequant<<<IN_F / 8, 256, 0, stream>>>(
        centroids, res_centroids, outlier_centroids, wscale, wbias,
        indices, res_indices, outlier_indices, perm, Wt);

    // Phase 2: out = x @ Wt   (8192 x 4096 x 4096, f32 WMMA)
    dim3 grid(OUT_F / BN, M_TOT / BM);
    vq_gemm<<<grid, 256, 0, stream>>>(x, Wt, out);
}